// DNC_46789373722625
// MI455X (gfx1250) — compile-verified
//
#include <hip/hip_runtime.h>
#include <hip/hip_bf16.h>
#include <math.h>

// ---------------- problem constants ----------------
#define N_SLOTS 256
#define WM_     64
#define R_      4
#define H_      512
#define D_      512
#define T_      128
#define B_      32
#define EPS_    1e-6f

// gates GEMV: K = D + R*WM + H = 1280, N = 4H = 2048
#define GK  1280
#define GN  2048
#define GKT (GK/4)      // 320 k-tiles
#define GNT (GN/16)     // 128 n-tiles
// iface GEMV: K = 512, N = 471 padded to 512 (32 n-tiles, mult of 4)
#define IK  512
#define INP 512
#define IKT (IK/4)      // 128
#define INT_ (INP/16)   // 32
// out GEMV: K = H + R*WM = 768, N = 512
#define OK  768
#define ON  512
#define OKT (OK/4)      // 192
#define ONT (ON/16)     // 32

// workspace layout (bytes)
#define WG_ELEMS ((size_t)GKT*GNT*64)
#define WI_ELEMS ((size_t)IKT*INT_*64)
#define WO_ELEMS ((size_t)OKT*ONT*64)
#define OFF_WG   ((size_t)0)
#define OFF_WI   (OFF_WG + WG_ELEMS*4)
#define OFF_WO   (OFF_WI + WI_ELEMS*4)
#define OFF_LINK (OFF_WO + WO_ELEMS*4)

typedef __attribute__((ext_vector_type(2))) float v2f;
typedef __attribute__((ext_vector_type(8))) float v8f;

#if defined(__HIP_DEVICE_COMPILE__) && __has_builtin(__builtin_amdgcn_wmma_f32_16x16x4_f32)
#define USE_WMMA_F32 1
#else
#define USE_WMMA_F32 0
#endif

#define DEV static __device__ __forceinline__

DEV float sigm(float x) { return 1.0f / (1.0f + __expf(-x)); }
DEV float softplus1(float x) {
  return 1.0f + ((x > 0.f) ? (x + log1pf(__expf(-x))) : log1pf(__expf(x)));
}

// ---------------- LDS layout (dynamic shared) ----------------
struct SmemLayout {
  float M[N_SLOTS * 65];       // memory matrix, stride 65 (bank-conflict pad)
  float H[H_];
  float C[H_];
  float gates[4 * H_];
  float z[INP];
  float rw[R_ * N_SLOTS];
  float fw[R_ * N_SLOTS];
  float bw[R_ * N_SLOTS];
  float cr[R_ * N_SLOTS];
  float usage[N_SLOTS];
  float ww[N_SLOTS];
  float prec[N_SLOTS];
  float allocw[N_SLOTS];
  float cw[N_SLOTS];
  float inv[N_SLOTS];
  float keys[5 * WM_];         // 4 read keys + write key, normalized
  float beta[8];               // beta[0..3] read, beta[4] write
  float freeg[4];
  float scal[4];               // 0: alloc_g, 1: write_g, 2: sum(ww)
  float erase[WM_];
  float wvec[WM_];
  float modes[12];             // [r][3]
  float sv[N_SLOTS];           // sort values
  float cp[N_SLOTS];           // cumprod scan
  float red[N_SLOTS];          // reduction scratch
  float rvec[R_ * WM_];
  float vin[GK];               // fp32 input vector for GEMVs
  int   si[N_SLOTS];           // sort indices
};

// ---------------- reductions (256 threads) ----------------
DEV float block_max(float* red, float v, int tid) {
  red[tid] = v; __syncthreads();
  for (int s = 128; s > 0; s >>= 1) {
    if (tid < s) red[tid] = fmaxf(red[tid], red[tid + s]);
    __syncthreads();
  }
  float r = red[0]; __syncthreads(); return r;
}
DEV float block_sum(float* red, float v, int tid) {
  red[tid] = v; __syncthreads();
  for (int s = 128; s > 0; s >>= 1) {
    if (tid < s) red[tid] += red[tid + s];
    __syncthreads();
  }
  float r = red[0]; __syncthreads(); return r;
}

// ---------------- fp32 WMMA GEMV ----------------
// Wp packed per (kt,nt) tile: [32 lanes][2 floats]; lane l holds
// B[k = kt*4 + (l>>4)*2 + q][n = nt*16 + (l&15)] for q=0,1.
// A-tile: row 0 = input vector chunk, rows 1..15 zero (via cndmask, no EXEC
// churn); D row 0 lives in acc[0] of lanes 0..15 per the 16x16 f32 layout.
// Processes 4 n-tiles per pass: one A build feeds 4 independent WMMA chains,
// and the 4 B loads sit at immediate offsets off one base address.
DEV void gemv_wmma(const float* __restrict__ Wp, const float* sV,
                   float* dst, const float* __restrict__ bias, int biasN,
                   int KT, int NT, int tid) {
#if USE_WMMA_F32
  const int lane  = tid & 31;
  const int wv    = tid >> 5;
  const bool row0 = ((lane & 15) == 0);
  const int ks    = (lane >> 4) * 2;
  for (int bi = wv; bi * 4 < NT; bi += 8) {
    const int nt0 = bi * 4;
    v8f acc0 = {}, acc1 = {}, acc2 = {}, acc3 = {};
    const float* wp = Wp + ((size_t)nt0 * 32 + lane) * 2;
    for (int kt = 0; kt < KT; ++kt) {
      float a0 = sV[kt * 4 + ks];
      float a1 = sV[kt * 4 + ks + 1];
      v2f a;
      a[0] = row0 ? a0 : 0.f;
      a[1] = row0 ? a1 : 0.f;
      const float* wk = wp + (size_t)kt * NT * 64;
      v2f b0 = *(const v2f*)(wk);
      v2f b1 = *(const v2f*)(wk + 64);
      v2f b2 = *(const v2f*)(wk + 128);
      v2f b3 = *(const v2f*)(wk + 192);
      acc0 = __builtin_amdgcn_wmma_f32_16x16x4_f32(false, a, false, b0, (short)0, acc0, false, false);
      acc1 = __builtin_amdgcn_wmma_f32_16x16x4_f32(false, a, false, b1, (short)0, acc1, false, false);
      acc2 = __builtin_amdgcn_wmma_f32_16x16x4_f32(false, a, false, b2, (short)0, acc2, false, false);
      acc3 = __builtin_amdgcn_wmma_f32_16x16x4_f32(false, a, false, b3, (short)0, acc3, false, false);
    }
    if (lane < 16) {
      int n = nt0 * 16 + lane;
      dst[n] = acc0[0] + ((n < biasN) ? bias[n] : 0.f);
      n += 16;
      dst[n] = acc1[0] + ((n < biasN) ? bias[n] : 0.f);
      n += 16;
      dst[n] = acc2[0] + ((n < biasN) ? bias[n] : 0.f);
      n += 16;
      dst[n] = acc3[0] + ((n < biasN) ? bias[n] : 0.f);
    }
  }
#else
  // scalar fallback (keeps compile green if the f32 WMMA builtin is absent)
  const int N = NT * 16, K = KT * 4;
  for (int n = tid; n < N; n += 256) {
    float acc = 0.f;
    int nt = n >> 4, nl = n & 15;
    for (int k = 0; k < K; ++k) {
      int kt = k >> 2, kl = k & 3;
      int lane = ((kl >> 1) << 4) | nl;
      size_t idx = (((size_t)kt * NT + nt) * 32 + lane) * 2 + (kl & 1);
      acc += sV[k] * Wp[idx];
    }
    dst[n] = acc + ((n < biasN) ? bias[n] : 0.f);
  }
#endif
}

// ---------------- weight packing ----------------
// dst tile layout matches gemv_wmma's B gather. Rows [0,Ksplit) from A,
// [Ksplit,K) from Bsrc; columns >= Nvalid are zero padding.
__global__ void pack_w(const float* __restrict__ A, const float* __restrict__ Bsrc,
                       float* __restrict__ dst, int KT, int NT, int Nvalid,
                       int Ksplit, int ld, int total) {
  int idx = blockIdx.x * blockDim.x + threadIdx.x;
  if (idx >= total) return;
  int q  = idx & 1;
  int l  = (idx >> 1) & 31;
  int t  = idx >> 6;
  int nt = t % NT;
  int kt = t / NT;
  int k  = kt * 4 + ((l >> 4) * 2) + q;
  int n  = nt * 16 + (l & 15);
  float v = 0.f;
  if (n < Nvalid)
    v = (k < Ksplit) ? A[(size_t)k * ld + n] : Bsrc[(size_t)(k - Ksplit) * ld + n];
  dst[idx] = v;
}

// ---------------- main scan kernel: 1 block = 1 batch element ----------------
__global__ __launch_bounds__(256) void dnc_scan(
    const float* __restrict__ emb,     // (T,B,D)
    const float* __restrict__ b_lstm,  // (2048)
    const float* __restrict__ b_ifc,   // (471)
    const float* __restrict__ b_pre,   // (512)
    const float* __restrict__ Wg,      // packed (1280 x 2048)
    const float* __restrict__ Wi,      // packed (512 x 512pad)
    const float* __restrict__ Wo,      // packed (768 x 512)
    float* __restrict__ linkAll,       // (B, 256, 256) in workspace
    float* __restrict__ out)           // (T,B,512)
{
  extern __shared__ char smem_raw[];
  SmemLayout& S = *(SmemLayout*)smem_raw;
  const int tid = threadIdx.x;
  const int b   = blockIdx.x;
  float* LG = linkAll + (size_t)b * (N_SLOTS * N_SLOTS);

  // ---- init carry ----
  for (int i = tid; i < H_; i += 256) { S.H[i] = 0.f; S.C[i] = 0.f; }
  S.usage[tid] = 0.f; S.ww[tid] = 0.f; S.prec[tid] = 0.f; S.rvec[tid] = 0.f;
  for (int r = 0; r < R_; ++r) S.rw[r * N_SLOTS + tid] = 0.f;
  for (int i = tid; i < N_SLOTS * WM_; i += 256) {
    int n = i >> 6, w = i & 63;
    S.M[n * 65 + w] = EPS_;
  }
  for (int i = tid; i < N_SLOTS * N_SLOTS; i += 256) LG[i] = 0.f;
  __syncthreads();

  for (int t = 0; t < T_; ++t) {
    // ---- build vin = [x(512), rvec(256), h(512)] ----
    const float* x = emb + ((size_t)t * B_ + b) * D_;
    for (int i = tid; i < D_; i += 256) S.vin[i] = x[i];
    S.vin[D_ + tid] = S.rvec[tid];
    for (int i = tid; i < H_; i += 256) S.vin[D_ + R_ * WM_ + i] = S.H[i];
    __syncthreads();

    // ---- gates = vin @ [Wx;Wh] + b_lstm ----
    gemv_wmma(Wg, S.vin, S.gates, b_lstm, 4 * H_, GKT, GNT, tid);
    __syncthreads();

    // ---- LSTM cell ----
    for (int i = tid; i < H_; i += 256) {
      float ig = sigm(S.gates[i]);
      float fg = sigm(S.gates[H_ + i]);
      float gg = tanhf(S.gates[2 * H_ + i]);
      float og = sigm(S.gates[3 * H_ + i]);
      float c  = fg * S.C[i] + ig * gg;
      S.C[i] = c;
      S.H[i] = og * tanhf(c);
    }
    __syncthreads();

    // ---- z = h @ W_iface + b_iface (vin[0..511] := new h) ----
    for (int i = tid; i < H_; i += 256) S.vin[i] = S.H[i];
    __syncthreads();
    gemv_wmma(Wi, S.vin, S.z, b_ifc, 471, IKT, INT_, tid);
    __syncthreads();

    // ---- parse interface vector ----
    if (tid < 4) {
      S.beta[tid]  = softplus1(S.z[256 + tid]);           // read strengths
      S.freeg[tid] = sigm(S.z[453 + tid]);                // free gates
      float e0 = S.z[459 + tid * 3], e1 = S.z[460 + tid * 3], e2 = S.z[461 + tid * 3];
      float mx = fmaxf(e0, fmaxf(e1, e2));
      float a0 = __expf(e0 - mx), a1 = __expf(e1 - mx), a2 = __expf(e2 - mx);
      float s = a0 + a1 + a2;
      S.modes[tid * 3 + 0] = a0 / s;
      S.modes[tid * 3 + 1] = a1 / s;
      S.modes[tid * 3 + 2] = a2 / s;
    } else if (tid == 4) {
      S.beta[4]  = softplus1(S.z[324]);                   // write strength
      S.scal[0]  = sigm(S.z[457]);                        // alloc gate
      S.scal[1]  = sigm(S.z[458]);                        // write gate
    }
    if (tid < WM_) {
      S.erase[tid] = sigm(S.z[325 + tid]);
      S.wvec[tid]  = S.z[389 + tid];
    }
    if (tid >= 64 && tid < 69) {                          // normalize 5 keys
      int kk = tid - 64;
      int base = (kk < 4) ? kk * 64 : 260;
      float ss = 0.f;
      for (int w = 0; w < WM_; ++w) { float v = S.z[base + w]; ss += v * v; }
      float invn = 1.0f / fmaxf(sqrtf(ss), EPS_);
      for (int w = 0; w < WM_; ++w) S.keys[kk * 64 + w] = S.z[base + w] * invn;
    }
    __syncthreads();

    // ---- usage update (old ww, old rw) ----
    {
      float ret = 1.f;
      #pragma unroll
      for (int r = 0; r < R_; ++r)
        ret *= (1.f - S.freeg[r] * S.rw[r * N_SLOTS + tid]);
      float u = S.usage[tid], w = S.ww[tid];
      S.usage[tid] = (u + w - u * w) * ret;
    }
    __syncthreads();

    // ---- allocation: bitonic sort (val,idx) ascending, stable-ish ----
    S.sv[tid] = S.usage[tid]; S.si[tid] = tid;
    __syncthreads();
    for (int k = 2; k <= N_SLOTS; k <<= 1) {
      for (int j = k >> 1; j > 0; j >>= 1) {
        int ixj = tid ^ j;
        if (ixj > tid) {
          bool up = ((tid & k) == 0);
          float va = S.sv[tid], vb = S.sv[ixj];
          int ia = S.si[tid], ib = S.si[ixj];
          bool agtb = (va > vb) || (va == vb && ia > ib);
          if (agtb == up) {
            S.sv[tid] = vb; S.sv[ixj] = va;
            S.si[tid] = ib; S.si[ixj] = ia;
          }
        }
        __syncthreads();
      }
    }
    // inclusive cumprod scan, then exclusive, scatter allocation
    S.cp[tid] = S.sv[tid];
    __syncthreads();
    for (int off = 1; off < N_SLOTS; off <<= 1) {
      float v = (tid >= off) ? S.cp[tid - off] : 1.f;
      __syncthreads();
      S.cp[tid] *= v;
      __syncthreads();
    }
    {
      float excl = (tid == 0) ? 1.f : S.cp[tid - 1];
      S.allocw[S.si[tid]] = (1.f - S.sv[tid]) * excl;
    }
    __syncthreads();

    // ---- write content weights (pre-update M) ----
    {
      float ss = 0.f;
      for (int w = 0; w < WM_; ++w) { float v = S.M[tid * 65 + w]; ss += v * v; }
      S.inv[tid] = 1.f / fmaxf(sqrtf(ss), EPS_);
    }
    {
      float dot = 0.f;
      for (int w = 0; w < WM_; ++w) dot += S.keys[4 * 64 + w] * S.M[tid * 65 + w];
      float sim = S.beta[4] * dot * S.inv[tid];
      float mx = block_max(S.red, sim, tid);
      float e  = __expf(sim - mx);
      float sm = block_sum(S.red, e, tid);
      S.cw[tid] = e / sm;
    }
    // ---- write weighting ----
    {
      float ag = S.scal[0], wg = S.scal[1];
      S.ww[tid] = wg * (ag * S.allocw[tid] + (1.f - ag) * S.cw[tid]);
    }
    __syncthreads();
    {
      float s = block_sum(S.red, S.ww[tid], tid);
      if (tid == 0) S.scal[2] = s;
    }
    __syncthreads();

    // ---- memory write ----
    {
      float w = S.ww[tid];
      for (int j = 0; j < WM_; ++j) {
        float m = S.M[tid * 65 + j];
        S.M[tid * 65 + j] = m * (1.f - w * S.erase[j]) + w * S.wvec[j];
      }
    }
    __syncthreads();

    // ---- read content weights (post-update M) ----
    {
      float ss = 0.f;
      for (int w = 0; w < WM_; ++w) { float v = S.M[tid * 65 + w]; ss += v * v; }
      S.inv[tid] = 1.f / fmaxf(sqrtf(ss), EPS_);
    }
    for (int r = 0; r < R_; ++r) {
      float dot = 0.f;
      for (int w = 0; w < WM_; ++w) dot += S.keys[r * 64 + w] * S.M[tid * 65 + w];
      float sim = S.beta[r] * dot * S.inv[tid];
      float mx = block_max(S.red, sim, tid);
      float e  = __expf(sim - mx);
      float sm = block_sum(S.red, e, tid);
      S.cr[r * N_SLOTS + tid] = e / sm;
    }
    __syncthreads();

    // ---- link pass A (thread = column j, coalesced): update link + bw ----
    {
      float bw0 = 0, bw1 = 0, bw2 = 0, bw3 = 0;
      float wj = S.ww[tid], pj = S.prec[tid];
      for (int i = 0; i < N_SLOTS; ++i) {
        float l  = LG[i * N_SLOTS + tid];
        float nl = (1.f - S.ww[i] - wj) * l + S.ww[i] * pj;
        if (i == tid) nl = 0.f;
        LG[i * N_SLOTS + tid] = nl;
        bw0 += nl * S.rw[0 * N_SLOTS + i];
        bw1 += nl * S.rw[1 * N_SLOTS + i];
        bw2 += nl * S.rw[2 * N_SLOTS + i];
        bw3 += nl * S.rw[3 * N_SLOTS + i];
      }
      S.bw[0 * N_SLOTS + tid] = bw0;
      S.bw[1 * N_SLOTS + tid] = bw1;
      S.bw[2 * N_SLOTS + tid] = bw2;
      S.bw[3 * N_SLOTS + tid] = bw3;
    }
    __threadfence_block();
    __syncthreads();
    // precedence update (uses old prec of own slot only)
    S.prec[tid] = (1.f - S.scal[2]) * S.prec[tid] + S.ww[tid];

    // ---- link pass B (thread = row i): fw from fresh link ----
    {
      float f0 = 0, f1 = 0, f2 = 0, f3 = 0;
      for (int j = 0; j < N_SLOTS; ++j) {
        float nl = LG[tid * N_SLOTS + j];
        f0 += nl * S.rw[0 * N_SLOTS + j];
        f1 += nl * S.rw[1 * N_SLOTS + j];
        f2 += nl * S.rw[2 * N_SLOTS + j];
        f3 += nl * S.rw[3 * N_SLOTS + j];
      }
      S.fw[0 * N_SLOTS + tid] = f0;
      S.fw[1 * N_SLOTS + tid] = f1;
      S.fw[2 * N_SLOTS + tid] = f2;
      S.fw[3 * N_SLOTS + tid] = f3;
    }
    __syncthreads();

    // ---- read weighting update ----
    #pragma unroll
    for (int r = 0; r < R_; ++r) {
      S.rw[r * N_SLOTS + tid] =
          S.modes[r * 3 + 0] * S.bw[r * N_SLOTS + tid] +
          S.modes[r * 3 + 1] * S.cr[r * N_SLOTS + tid] +
          S.modes[r * 3 + 2] * S.fw[r * N_SLOTS + tid];
    }
    __syncthreads();

    // ---- read vectors: rvec[r][w] = sum_n rw[r][n] * M[n][w] ----
    {
      int r = tid >> 6, w = tid & 63;
      float s = 0.f;
      for (int n = 0; n < N_SLOTS; ++n)
        s += S.rw[r * N_SLOTS + n] * S.M[n * 65 + w];
      S.rvec[tid] = s;
    }
    __syncthreads();

    // ---- out = [h; rvec] @ [W_pre; W_rout] + b_pre ----
    S.vin[H_ + tid] = S.rvec[tid];   // vin[0..511] already holds new h
    __syncthreads();
    gemv_wmma(Wo, S.vin, out + ((size_t)t * B_ + b) * D_, b_pre, ON, OKT, ONT, tid);
    __syncthreads();
  }
}

// ---------------- host entry ----------------
extern "C" void kernel_launch(void* const* d_in, const int* in_sizes, int n_in,
                              void* d_out, int out_size, void* d_ws, size_t ws_size,
                              hipStream_t stream) {
  (void)in_sizes; (void)n_in; (void)out_size; (void)ws_size;
  const float* emb    = (const float*)d_in[0];  // (T,B,D)
  const float* Wx     = (const float*)d_in[1];  // (768,2048)
  const float* Wh     = (const float*)d_in[2];  // (512,2048)
  const float* b_lstm = (const float*)d_in[3];  // (2048)
  const float* W_pre  = (const float*)d_in[4];  // (512,512)
  const float* b_pre  = (const float*)d_in[5];  // (512)
  const float* W_ifc  = (const float*)d_in[6];  // (512,471)
  const float* b_ifc  = (const float*)d_in[7];  // (471)
  const float* W_rout = (const float*)d_in[8];  // (256,512)

  char* ws = (char*)d_ws;
  float* Wg   = (float*)(ws + OFF_WG);
  float* Wi   = (float*)(ws + OFF_WI);
  float* Wo   = (float*)(ws + OFF_WO);
  float* link = (float*)(ws + OFF_LINK);

  // pack weights into WMMA tile-major layouts
  {
    int tot = (int)WG_ELEMS;
    pack_w<<<(tot + 255) / 256, 256, 0, stream>>>(Wx, Wh, Wg, GKT, GNT, GN, 768, GN, tot);
  }
  {
    int tot = (int)WI_ELEMS;
    pack_w<<<(tot + 255) / 256, 256, 0, stream>>>(W_ifc, W_ifc, Wi, IKT, INT_, 471, IK, 471, tot);
  }
  {
    int tot = (int)WO_ELEMS;
    pack_w<<<(tot + 255) / 256, 256, 0, stream>>>(W_pre, W_rout, Wo, OKT, ONT, ON, 512, ON, tot);
  }

  int smem = (int)sizeof(SmemLayout);
  (void)hipFuncSetAttribute(reinterpret_cast<const void*>(&dnc_scan),
                            hipFuncAttributeMaxDynamicSharedMemorySize, smem);
  dnc_scan<<<B_, 256, smem, stream>>>(emb, b_lstm, b_ifc, b_pre,
                                      Wg, Wi, Wo, link, (float*)d_out);
}